// VariableGroupNorm_4363686773082
// MI455X (gfx1250) — compile-verified
//
#include <hip/hip_runtime.h>

typedef __attribute__((ext_vector_type(2))) float v2f;
typedef __attribute__((ext_vector_type(4))) float v4f;
typedef __attribute__((ext_vector_type(8))) float v8f;

#define NB   64      // batch
#define CB   256     // channels
#define HWN  3136    // 56*56
#define HW4  784     // HWN/4, exact
#define GNUM 8       // number of groups
#define EPSF 1e-12f

// ---------------------------------------------------------------------------
// Stage 1: per-(n,c) spatial sum & sum-of-squares. One block per channel.
// Streaming b128 loads (regular temporal -> allocate in 192MB L2 so stage 3
// can re-hit them), wave32 shuffle reduce, tiny LDS combine.
// ---------------------------------------------------------------------------
__global__ __launch_bounds__(256) void vgn_chan_reduce(const float* __restrict__ x,
                                                       float* __restrict__ ch_sum,
                                                       float* __restrict__ ch_sq) {
  const int nc = blockIdx.x;                       // nc = n*CB + c
  const v4f* __restrict__ xp =
      reinterpret_cast<const v4f*>(x) + (size_t)nc * HW4;
  float s = 0.f, q = 0.f;
  for (int i = threadIdx.x; i < HW4; i += 256) {
    __builtin_prefetch((const void*)(xp + i + 512), 0, 0);  // global_prefetch_b8
    v4f v = xp[i];
    s += (v.x + v.y) + (v.z + v.w);
    q += (v.x * v.x + v.y * v.y) + (v.z * v.z + v.w * v.w);
  }
  for (int off = 16; off > 0; off >>= 1) {
    s += __shfl_down(s, off, 32);
    q += __shfl_down(q, off, 32);
  }
  __shared__ float ss[8], sq[8];
  const int wave = threadIdx.x >> 5;
  if ((threadIdx.x & 31) == 0) { ss[wave] = s; sq[wave] = q; }
  __syncthreads();
  if (threadIdx.x == 0) {
    float ts = 0.f, tq = 0.f;
#pragma unroll
    for (int w2 = 0; w2 < 8; ++w2) { ts += ss[w2]; tq += sq[w2]; }
    ch_sum[nc] = ts;
    ch_sq[nc]  = tq;
  }
}

// ---------------------------------------------------------------------------
// Stage 2: segment-sum over channels as a WMMA f32 GEMM.
//   gsum[G,N] = Member[G(pad16),C=256] x ch_sum^T[C,N=64]
// 1 block, 4 waves; wave w owns the 16x16 output tile for n in [16w,16w+16).
// Two accumulator chains (sum, sqsum) share the A (membership) fragment.
// A layout (32-bit 16x4 A): lanes0-15 M=0..15; v.x = K{0|2}, v.y = K{1|3}.
// D layout: VGPR r -> row r + 8*(lane>=16), col = lane&15.
// ---------------------------------------------------------------------------
__global__ __launch_bounds__(128) void vgn_group_stats(const float* __restrict__ ch_sum,
                                                       const float* __restrict__ ch_sq,
                                                       const int* __restrict__ gsz,
                                                       float* __restrict__ mu,
                                                       float* __restrict__ ivar) {
  const int lane = threadIdx.x & 31;
  const int wave = threadIdx.x >> 5;
  const int n0   = wave * 16;

  int pre[GNUM + 1];
  pre[0] = 0;
#pragma unroll
  for (int g = 0; g < GNUM; ++g) pre[g + 1] = pre[g] + gsz[g];

  const int mrow  = lane & 15;   // padded group row (A matrix M)
  const int khalf = lane >> 4;   // 0 -> K{k0,k0+1}; 1 -> K{k0+2,k0+3}
  const int ncol  = lane & 15;   // B matrix column (batch index offset)

  v8f acc_s = {0.f, 0.f, 0.f, 0.f, 0.f, 0.f, 0.f, 0.f};
  v8f acc_q = acc_s;

  for (int k0 = 0; k0 < CB; k0 += 4) {
    const int ka = k0 + khalf * 2;
    int g0 = 0, g1 = 0;                 // group id of channels ka, ka+1
#pragma unroll
    for (int k = 1; k < GNUM; ++k) {
      if (ka     >= pre[k]) g0 = k;
      if (ka + 1 >= pre[k]) g1 = k;
    }
    v2f a;
    a.x = (mrow < GNUM && g0 == mrow) ? 1.f : 0.f;
    a.y = (mrow < GNUM && g1 == mrow) ? 1.f : 0.f;

    const float* bs_p = ch_sum + (size_t)(n0 + ncol) * CB + ka;
    const float* bq_p = ch_sq  + (size_t)(n0 + ncol) * CB + ka;
    v2f bs; bs.x = bs_p[0]; bs.y = bs_p[1];
    v2f bq; bq.x = bq_p[0]; bq.y = bq_p[1];

    acc_s = __builtin_amdgcn_wmma_f32_16x16x4_f32(false, a, false, bs,
                                                  (short)0, acc_s, false, false);
    acc_q = __builtin_amdgcn_wmma_f32_16x16x4_f32(false, a, false, bq,
                                                  (short)0, acc_q, false, false);
  }

  const int colD  = lane & 15;
  const int rbase = (lane >> 4) * 8;
#pragma unroll
  for (int r = 0; r < 8; ++r) {
    const int g = rbase + r;
    if (g < GNUM) {
      // v_rcp_f32 instead of the full IEEE divide chain (stats precision)
      const float rcnt = __builtin_amdgcn_rcpf((float)gsz[g] * (float)HWN);
      const float m    = acc_s[r] * rcnt;
      const float v    = acc_q[r] * rcnt - m * m;
      const float iv   = rsqrtf(v + EPSF);
      mu[g * NB + n0 + colD]   = m;
      ivar[g * NB + n0 + colD] = iv;
    }
  }
}

// ---------------------------------------------------------------------------
// Stage 3: normalize. Reverse block order: the tail of x (written through L2
// by stage 1 most recently) is still resident in the 192MB L2, so walking
// backwards re-hits ~93% of x. Output uses non-temporal stores so the write
// stream does not evict x from L2.
// ---------------------------------------------------------------------------
__global__ __launch_bounds__(256) void vgn_normalize(const float* __restrict__ x,
                                                     const float* __restrict__ weight,
                                                     const float* __restrict__ bias,
                                                     const int* __restrict__ gsz,
                                                     const float* __restrict__ mu,
                                                     const float* __restrict__ ivar,
                                                     float* __restrict__ out) {
  const int nc = (int)gridDim.x - 1 - (int)blockIdx.x;   // reversed traversal
  const int n  = nc >> 8;          // CB == 256
  const int c  = nc & (CB - 1);

  int g = 0, p = 0;
#pragma unroll
  for (int k = 0; k < GNUM; ++k) {
    p += gsz[k];
    if (c >= p) g = k + 1;
  }

  const float m     = mu[g * NB + n];
  const float iv    = ivar[g * NB + n];
  const float scale = iv * weight[c];
  const float shift = bias[c] - m * scale;

  const v4f* __restrict__ xp = reinterpret_cast<const v4f*>(x) + (size_t)nc * HW4;
  v4f* __restrict__ op       = reinterpret_cast<v4f*>(out) + (size_t)nc * HW4;

  for (int i = threadIdx.x; i < HW4; i += 256) {
    v4f v = xp[i];                       // RT load: expect L2 hit
    v4f r;
    r.x = v.x * scale + shift;
    r.y = v.y * scale + shift;
    r.z = v.z * scale + shift;
    r.w = v.w * scale + shift;
    __builtin_nontemporal_store(r, op + i);   // TH=NT store: bypass L2 LRU
  }
}

// ---------------------------------------------------------------------------
extern "C" void kernel_launch(void* const* d_in, const int* in_sizes, int n_in,
                              void* d_out, int out_size, void* d_ws, size_t ws_size,
                              hipStream_t stream) {
  (void)in_sizes; (void)n_in; (void)out_size; (void)ws_size;
  const float* x      = (const float*)d_in[0];
  const float* weight = (const float*)d_in[1];
  const float* bias   = (const float*)d_in[2];
  const int*   gsz    = (const int*)d_in[3];
  float*       out    = (float*)d_out;

  float* ws      = (float*)d_ws;
  float* ch_sum  = ws;                         // NB*CB floats
  float* ch_sq   = ws + NB * CB;               // NB*CB floats
  float* mu      = ws + 2 * NB * CB;           // GNUM*NB floats
  float* ivar    = ws + 2 * NB * CB + GNUM * NB;

  vgn_chan_reduce<<<NB * CB, 256, 0, stream>>>(x, ch_sum, ch_sq);
  vgn_group_stats<<<1, 128, 0, stream>>>(ch_sum, ch_sq, gsz, mu, ivar);
  vgn_normalize<<<NB * CB, 256, 0, stream>>>(x, weight, bias, gsz, mu, ivar, out);
}